// SamplerQNN_65481071397000
// MI455X (gfx1250) — compile-verified
//
#include <hip/hip_runtime.h>

// SamplerQNN ring circuit, solved analytically:
//   z_v      = cos(x_v + theta_v)
//   out[b,w] = prod_{v=0..w} z_v      (w = 1..9)
//   out[b,0] = prod_{v=1..9} z_v
// Memory-bound streaming kernel. Global traffic is staged through LDS with the
// CDNA5 async global<->LDS engine (ASYNCcnt) in GVS (saddr + 32-bit voffset)
// form so per-lane address math stays 32-bit.

#define NQ   10
#define BLK  256
#define TILE (BLK * NQ)   // 2560 floats = 10240 bytes per block

__global__ __launch_bounds__(BLK) void qnn_ring_z_kernel(
    const float* __restrict__ x, const float* __restrict__ theta,
    float* __restrict__ out, int nElems /* batch * NQ, fits in i32 */) {
  __shared__ float tin[TILE];
  __shared__ float tout[TILE];

  const int t = (int)threadIdx.x;
  const int eBase = (int)blockIdx.x * TILE;       // element index of tile start

  // theta is tiny and uniform -> scalar-cache broadcast loads.
  float th[NQ];
#pragma unroll
  for (int v = 0; v < NQ; ++v) th[v] = theta[v];

  const unsigned ldsIn  = (unsigned)(size_t)(&tin[0]);   // low 32 bits = LDS offset
  const unsigned ldsOut = (unsigned)(size_t)(&tout[0]);

  // ---- Stage x tile into LDS: 10 fully-coalesced async b32 streams/wave ----
  // GVS mode: addr = SGPR_base + 32-bit VGPR byte offset. Clamp = one v_min_i32.
#pragma unroll
  for (int k = 0; k < NQ; ++k) {
    int e = eBase + k * BLK + t;
    e = min(e, nElems - 1);            // tail clamp (blocks are full for BATCH=65536)
    unsigned lo   = ldsIn + (unsigned)((k * BLK + t) * 4);
    unsigned voff = (unsigned)e * 4u;  // byte offset from base
    asm volatile("global_load_async_to_lds_b32 %0, %1, %2"
                 :: "v"(lo), "v"(voff), "s"(x) : "memory");
  }
  asm volatile("s_wait_asynccnt 0" ::: "memory");  // per-wave ASYNCcnt drain
  __syncthreads();                                  // tile visible to all waves

  // ---- Per-row analytic evaluation (one batch row per thread) ----
  float z[NQ];
#pragma unroll
  for (int v = 0; v < NQ; ++v)
    z[v] = cosf(tin[t * NQ + v] + th[v]);

  float o[NQ];
  float p = z[0];
#pragma unroll
  for (int w = 1; w < NQ; ++w) { p *= z[w]; o[w] = p; }   // prefix products
  float q = z[1];
#pragma unroll
  for (int v = 2; v < NQ; ++v) q *= z[v];                 // suffix for wire 0
  o[0] = q;

#pragma unroll
  for (int v = 0; v < NQ; ++v) tout[t * NQ + v] = o[v];
  __syncthreads();

  // ---- Coalesced async LDS->global store of the output tile ----
#pragma unroll
  for (int k = 0; k < NQ; ++k) {
    int e = eBase + k * BLK + t;
    if (e < nElems) {   // EXEC-predicated async transfer on tail (never taken here)
      unsigned lo   = ldsOut + (unsigned)((k * BLK + t) * 4);
      unsigned voff = (unsigned)e * 4u;
      asm volatile("global_store_async_from_lds_b32 %0, %1, %2"
                   :: "v"(voff), "v"(lo), "s"(out) : "memory");
    }
  }
  asm volatile("s_wait_asynccnt 0" ::: "memory");
}

extern "C" void kernel_launch(void* const* d_in, const int* in_sizes, int n_in,
                              void* d_out, int out_size, void* d_ws, size_t ws_size,
                              hipStream_t stream) {
  (void)n_in; (void)out_size; (void)d_ws; (void)ws_size;
  const float* x     = (const float*)d_in[0];   // (BATCH, 10) float32
  const float* theta = (const float*)d_in[1];   // (10,)     float32
  float* out = (float*)d_out;                   // (BATCH, 10) float32

  const int nElems  = in_sizes[0];                    // BATCH * NQ
  const int nBlocks = (nElems + TILE - 1) / TILE;     // 256 for BATCH=65536
  qnn_ring_z_kernel<<<dim3(nBlocks), dim3(BLK), 0, stream>>>(x, theta, out, nElems);
}